// NnHalfKP_29566554865845
// MI455X (gfx1250) — compile-verified
//
#include <hip/hip_runtime.h>
#include <math.h>

typedef float v2f __attribute__((ext_vector_type(2)));
typedef float v8f __attribute__((ext_vector_type(8)));

#define FT_OUT 256
#define V_IN   640

// ---------------- helpers ----------------
__device__ __forceinline__ float4 f4add(float4 a, float4 b) {
  return make_float4(a.x + b.x, a.y + b.y, a.z + b.z, a.w + b.w);
}
__device__ __forceinline__ float4 f4fma(float s, float4 a, float4 acc) {
  return make_float4(fmaf(s, a.x, acc.x), fmaf(s, a.y, acc.y),
                     fmaf(s, a.z, acc.z), fmaf(s, a.w, acc.w));
}
__device__ __forceinline__ float4 f4clip01(float4 a) {
  return make_float4(fminf(fmaxf(a.x, 0.f), 1.f), fminf(fmaxf(a.y, 0.f), 1.f),
                     fminf(fmaxf(a.z, 0.f), 1.f), fminf(fmaxf(a.w, 0.f), 1.f));
}
__device__ __forceinline__ float f4dot(float4 a, float4 b) {
  return fmaf(a.x, b.x, fmaf(a.y, b.y, fmaf(a.z, b.z, a.w * b.w)));
}

// ---------------- 1) tiled transpose: out[c][r] = in[r][c] ----------------
__global__ void k_transpose(const float* __restrict__ in, float* __restrict__ out,
                            int R, int C) {
  __shared__ float tile[32][33];
  const int x0 = blockIdx.x * 32, y0 = blockIdx.y * 32;
  const int tx = threadIdx.x, ty = threadIdx.y;       // 32 x 8
#pragma unroll
  for (int i = 0; i < 32; i += 8)
    tile[ty + i][tx] = in[(size_t)(y0 + ty + i) * C + (x0 + tx)];
  __syncthreads();
#pragma unroll
  for (int i = 0; i < 32; i += 8)
    out[(size_t)(x0 + ty + i) * R + (y0 + tx)] = tile[tx][ty + i];
}

// ---------------- 2) densify folded-fft sparse rows: S[2B][640] ----------------
__global__ void k_densify(const int* __restrict__ sIdx, const int* __restrict__ nIdx,
                          const float* __restrict__ vals, float* __restrict__ S,
                          int B, int fpp) {
  __shared__ float lds[8 * V_IN];                     // 20 KB, one row per wave
  const int wslot = threadIdx.x >> 5;
  const int lane  = threadIdx.x & 31;
  const int g = blockIdx.x * 8 + wslot;               // [0, 2B)
  if (g >= 2 * B) return;
  float* row = lds + wslot * V_IN;
  for (int i = lane; i < V_IN; i += 32) row[i] = 0.0f;
  const int* idx = (g < B) ? sIdx : nIdx;
  const int  pos = (g < B) ? g : g - B;
  for (int i = lane; i < fpp; i += 32) {
    const int   f = idx[((size_t)pos * fpp + i) * 2 + 1];
    const float v = vals[(size_t)pos * fpp + i];
    atomicAdd(&row[((unsigned)f) % V_IN], v);         // ds_add_f32
  }
  __builtin_amdgcn_wave_barrier();
  float4*       dst = (float4*)(S + (size_t)g * V_IN);
  const float4* src = (const float4*)row;
  for (int i = lane; i < V_IN / 4; i += 32) dst[i] = src[i];
}

// ---------------- 3) small = S[2B x 640] @ Wt[640 x 256]  (f32 WMMA) ----------
// Per wave: 64x16 output tile; A layout: lanes0-15 K0/K1, lanes16-31 K2/K3;
// B layout (4x16): VGPR0 = {lanes0-15:K0, lanes16-31:K2}, VGPR1 = {K1, K3}.
__global__ void k_gemm_small(const float* __restrict__ S, const float* __restrict__ Wt,
                             float* __restrict__ C, int rows) {
  const int wid  = (blockIdx.x * blockDim.x + threadIdx.x) >> 5;
  const int lane = threadIdx.x & 31;
  const int NT   = FT_OUT / 16;                       // 16 col tiles
  const int tm = wid / NT, tn = wid % NT;
  const int row0 = tm * 64;
  if (row0 >= rows) return;
  const int lm = lane & 15, lk = lane >> 4;           // lk: 0/1 -> K offset 0/2
  const int col = tn * 16 + lm;
  v8f acc0 = {}, acc1 = {}, acc2 = {}, acc3 = {};
#pragma unroll 2
  for (int k0 = 0; k0 < V_IN; k0 += 4) {
    v2f b;
    b.x = Wt[(size_t)(k0 + lk * 2 + 0) * FT_OUT + col];
    b.y = Wt[(size_t)(k0 + lk * 2 + 1) * FT_OUT + col];
    const float* sb = S + (size_t)row0 * V_IN + k0 + lk * 2;
    v2f a0 = *(const v2f*)(sb + (size_t)(lm +  0) * V_IN);
    v2f a1 = *(const v2f*)(sb + (size_t)(lm + 16) * V_IN);
    v2f a2 = *(const v2f*)(sb + (size_t)(lm + 32) * V_IN);
    v2f a3 = *(const v2f*)(sb + (size_t)(lm + 48) * V_IN);
    acc0 = __builtin_amdgcn_wmma_f32_16x16x4_f32(false, a0, false, b, (short)0, acc0, false, false);
    acc1 = __builtin_amdgcn_wmma_f32_16x16x4_f32(false, a1, false, b, (short)0, acc1, false, false);
    acc2 = __builtin_amdgcn_wmma_f32_16x16x4_f32(false, a2, false, b, (short)0, acc2, false, false);
    acc3 = __builtin_amdgcn_wmma_f32_16x16x4_f32(false, a3, false, b, (short)0, acc3, false, false);
  }
#pragma unroll
  for (int j = 0; j < 8; ++j) {                       // D: VGPR j = rows j / j+8
    const int r = row0 + j + lk * 8;
    C[(size_t)(r +  0) * FT_OUT + col] = acc0[j];
    C[(size_t)(r + 16) * FT_OUT + col] = acc1[j];
    C[(size_t)(r + 32) * FT_OUT + col] = acc2[j];
    C[(size_t)(r + 48) * FT_OUT + col] = acc3[j];
  }
}

// ---------------- 4) per-position eval (one wave32 per position) --------------
__device__ float branch_dot(const int* __restrict__ idx, const float* __restrict__ vals,
                            int p, int fpp, int lane,
                            const float* __restrict__ ftBig, int ftT, int ftIn,
                            const float* __restrict__ fftW, const float* __restrict__ smallRow,
                            float4 bias0, float4 bias1, float4 ow0, float4 ow1) {
  float4 a0 = bias0, a1 = bias1;
  if (smallRow) {
    const float4* sm = (const float4*)smallRow;
    a0 = f4add(a0, sm[lane]);
    a1 = f4add(a1, sm[32 + lane]);
  }
  for (int kb = 0; kb < fpp; kb += 32) {
    int myF = 0; float myV = 0.f;
    const int i = kb + lane;
    if (i < fpp) {
      myF = idx[((size_t)p * fpp + i) * 2 + 1];
      myV = vals[(size_t)p * fpp + i];
      if (ftT) __builtin_prefetch(ftBig + (size_t)myF * FT_OUT, 0, 3);
    }
    const int kend = (fpp - kb < 32) ? (fpp - kb) : 32;
    for (int k = 0; k < kend; ++k) {
      const int   f = __shfl(myF, k);
      const float v = __shfl(myV, k);
      if (ftT) {
        const float4* r = (const float4*)(ftBig + (size_t)f * FT_OUT);
        a0 = f4fma(v, r[lane], a0);                   // channels lane*4..+3
        a1 = f4fma(v, r[32 + lane], a1);              // channels 128+lane*4..+3
      } else {                                        // strided fallback
        const float* c = ftBig + f;
        a0.x = fmaf(v, c[(size_t)(lane * 4 + 0) * ftIn], a0.x);
        a0.y = fmaf(v, c[(size_t)(lane * 4 + 1) * ftIn], a0.y);
        a0.z = fmaf(v, c[(size_t)(lane * 4 + 2) * ftIn], a0.z);
        a0.w = fmaf(v, c[(size_t)(lane * 4 + 3) * ftIn], a0.w);
        a1.x = fmaf(v, c[(size_t)(128 + lane * 4 + 0) * ftIn], a1.x);
        a1.y = fmaf(v, c[(size_t)(128 + lane * 4 + 1) * ftIn], a1.y);
        a1.z = fmaf(v, c[(size_t)(128 + lane * 4 + 2) * ftIn], a1.z);
        a1.w = fmaf(v, c[(size_t)(128 + lane * 4 + 3) * ftIn], a1.w);
      }
      if (!smallRow) {                                // fft gather fallback
        const float* fc = fftW + (((unsigned)f) % V_IN);
        a0.x = fmaf(v, fc[(size_t)(lane * 4 + 0) * V_IN], a0.x);
        a0.y = fmaf(v, fc[(size_t)(lane * 4 + 1) * V_IN], a0.y);
        a0.z = fmaf(v, fc[(size_t)(lane * 4 + 2) * V_IN], a0.z);
        a0.w = fmaf(v, fc[(size_t)(lane * 4 + 3) * V_IN], a0.w);
        a1.x = fmaf(v, fc[(size_t)(128 + lane * 4 + 0) * V_IN], a1.x);
        a1.y = fmaf(v, fc[(size_t)(128 + lane * 4 + 1) * V_IN], a1.y);
        a1.z = fmaf(v, fc[(size_t)(128 + lane * 4 + 2) * V_IN], a1.z);
        a1.w = fmaf(v, fc[(size_t)(128 + lane * 4 + 3) * V_IN], a1.w);
      }
    }
  }
  a0 = f4clip01(a0);
  a1 = f4clip01(a1);
  return f4dot(a0, ow0) + f4dot(a1, ow1);
}

__global__ void k_eval(const int* __restrict__ sIdx, const int* __restrict__ nIdx,
                       const float* __restrict__ vals,
                       const float* __restrict__ ftBig, int ftT, int ftIn,
                       const float* __restrict__ fftW, const float* __restrict__ smallBuf,
                       const float* __restrict__ ftb, const float* __restrict__ fftb,
                       const float* __restrict__ outW, const float* __restrict__ outb,
                       float* __restrict__ l1sig, int B, int fpp) {
  const int lane = threadIdx.x & 31;
  const int p = blockIdx.x * (blockDim.x >> 5) + (threadIdx.x >> 5);
  if (p >= B) return;
  const float4* ftb4  = (const float4*)ftb;
  const float4* fftb4 = (const float4*)fftb;
  const float4  bias0 = f4add(ftb4[lane], fftb4[lane]);
  const float4  bias1 = f4add(ftb4[32 + lane], fftb4[32 + lane]);
  const float4* ow = (const float4*)outW;
  float part = branch_dot(sIdx, vals, p, fpp, lane, ftBig, ftT, ftIn, fftW,
                          smallBuf ? smallBuf + (size_t)p * FT_OUT : nullptr,
                          bias0, bias1, ow[lane], ow[32 + lane]);
  part      += branch_dot(nIdx, vals, p, fpp, lane, ftBig, ftT, ftIn, fftW,
                          smallBuf ? smallBuf + (size_t)(B + p) * FT_OUT : nullptr,
                          bias0, bias1, ow[64 + lane], ow[96 + lane]);
  for (int off = 16; off > 0; off >>= 1) part += __shfl_down(part, off);
  if (lane == 0) l1sig[p] = 1.0f / (1.0f + expf(-(part + outb[0])));
}

// ---------------- 5) bucket indirection ----------------
__global__ void k_bucket(const float* __restrict__ l1sig, const int* __restrict__ buckets,
                         float* __restrict__ out, int B) {
  const int i = blockIdx.x * blockDim.x + threadIdx.x;
  if (i < B) out[i] = l1sig[buckets[i] + i];
}

// ---------------- launcher ----------------
extern "C" void kernel_launch(void* const* d_in, const int* in_sizes, int n_in,
                              void* d_out, int out_size, void* d_ws, size_t ws_size,
                              hipStream_t stream) {
  const int*   sIdx = (const int*)d_in[0];
  const int*   nIdx = (const int*)d_in[1];
  const float* vals = (const float*)d_in[2];
  const int*   bkts = (const int*)d_in[3];
  const float* ftW  = (const float*)d_in[4];
  const float* ftb  = (const float*)d_in[5];
  const float* fftW = (const float*)d_in[6];
  const float* fftb = (const float*)d_in[7];
  const float* outW = (const float*)d_in[8];
  const float* outb = (const float*)d_in[9];

  const int B    = in_sizes[3];
  const int nnz  = in_sizes[2];
  const int fpp  = nnz / B;                 // 32
  const int ftIn = in_sizes[4] / FT_OUT;    // 40960

  const size_t ftWtB  = (size_t)ftIn * FT_OUT * 4;
  const size_t fftWtB = (size_t)V_IN * FT_OUT * 4;
  const size_t SB     = (size_t)2 * B * V_IN * 4;
  const size_t smallB = (size_t)2 * B * FT_OUT * 4;
  const size_t l1B    = (size_t)B * 4;
  const size_t needFull = ftWtB + fftWtB + SB + smallB + l1B;

  char* ws = (char*)d_ws;
  const bool full = (ws_size >= needFull) && (ftIn % 32 == 0) && ((2 * B) % 64 == 0);

  float* ftWt   = (float*)ws;
  float* fftWt  = (float*)(ws + ftWtB);
  float* S      = (float*)(ws + ftWtB + fftWtB);
  float* smallM = (float*)(ws + ftWtB + fftWtB + SB);
  float* l1     = full ? (float*)(ws + ftWtB + fftWtB + SB + smallB)
                       : (ws_size >= l1B ? (float*)ws : nullptr);
  float* outF   = (float*)d_out;
  float* evalDst = l1 ? l1 : outF;

  if (full) {
    dim3 blk(32, 8);
    k_transpose<<<dim3(ftIn / 32, FT_OUT / 32), blk, 0, stream>>>(ftW, ftWt, FT_OUT, ftIn);
    k_transpose<<<dim3(V_IN / 32, FT_OUT / 32), blk, 0, stream>>>(fftW, fftWt, FT_OUT, V_IN);
    k_densify<<<(2 * B + 7) / 8, 256, 0, stream>>>(sIdx, nIdx, vals, S, B, fpp);
    const int totalWaves = ((2 * B) / 64) * (FT_OUT / 16);
    k_gemm_small<<<(totalWaves + 7) / 8, 256, 0, stream>>>(S, fftWt, smallM, 2 * B);
    k_eval<<<(B + 7) / 8, 256, 0, stream>>>(sIdx, nIdx, vals, ftWt, 1, ftIn, fftW, smallM,
                                            ftb, fftb, outW, outb, evalDst, B, fpp);
  } else {
    k_eval<<<(B + 7) / 8, 256, 0, stream>>>(sIdx, nIdx, vals, ftW, 0, ftIn, fftW, nullptr,
                                            ftb, fftb, outW, outb, evalDst, B, fpp);
  }
  if (l1) k_bucket<<<(B + 255) / 256, 256, 0, stream>>>(l1, bkts, outF, B);
}